// BktModel_16527034155050
// MI455X (gfx1250) — compile-verified
//
#include <hip/hip_runtime.h>

#define BB 256
#define TT 500
#define NKC 100
#define NA 13

typedef __attribute__((ext_vector_type(2))) float v2f;
typedef __attribute__((ext_vector_type(8))) float v8f;

__device__ __forceinline__ float lse2(float x, float y) {
    float m = fmaxf(x, y);
    return m + log1pf(__expf(-fabsf(x - y)));
}

// ---------------------------------------------------------------------------
// Kernel 1: x[B*T, 2] = FM[B*T, 200] * lr_w[2,200]^T + lr_b  via fp32 WMMA.
// One wave per 16-row tile; K=200 in 50 chunks of 4; N padded 2 -> 16.
// A layout (16x4 f32): lane L holds M=L&15, VGPR0/1 = K = (L<16 ? 0,1 : 2,3).
// B layout (4x16 f32): VGPR0/1 lanes 0-15 = rows K=0,1; lanes 16-31 = K=2,3.
// C/D layout: VGPR r -> row r (lanes 0-15) / row r+8 (lanes 16-31), N = lane&15.
// ---------------------------------------------------------------------------
__global__ __launch_bounds__(128)
void gemm_x_wmma(const float* __restrict__ FM, const float* __restrict__ lrw,
                 const float* __restrict__ lrb, float* __restrict__ xout) {
    __shared__ v2f bstage[50][32];
    const int tid = threadIdx.x;
    for (int i = tid; i < 50 * 32; i += 128) {
        int k = i >> 5, L = i & 31;
        int n = L & 15;
        int kb = k * 4 + ((L >> 4) << 1);
        v2f v;
        v.x = (n < 2) ? lrw[n * 200 + kb] : 0.0f;
        v.y = (n < 2) ? lrw[n * 200 + kb + 1] : 0.0f;
        bstage[k][L] = v;
    }
    __syncthreads();

    const int wave = tid >> 5, lane = tid & 31;
    const int tile = blockIdx.x * 4 + wave;          // 8000 tiles total
    const long row0 = (long)tile * 16;
    const int m = lane & 15;
    const int khalf = (lane >> 4) << 1;              // 0 or 2
    const float* arow = FM + (row0 + m) * 200 + khalf;

    v8f acc = {};
    for (int k = 0; k < 50; ++k) {
        v2f a = *(const v2f*)(arow + k * 4);
        v2f b = bstage[k][lane];
        acc = __builtin_amdgcn_wmma_f32_16x16x4_f32(
            false, a, false, b, (short)0, acc, false, false);
    }

    const int n = lane & 15;
    if (n < 2) {
        const float bias = lrb[n];
        const int moff = (lane >> 4) << 3;           // 0 or 8
        for (int r = 0; r < 8; ++r) {
            long row = row0 + moff + r;
            xout[row * 2 + n] = acc[r] + bias;
        }
    }
}

// ---------------------------------------------------------------------------
// Kernel 2: sequential HMM scan. One wave32 per sequence, lane = ability a.
// log_alpha[100][2][13] per sequence lives in LDS; log_t / la0 / obs_kc
// tables normalized once into LDS; obs_logits_problem gathered from L2.
// ---------------------------------------------------------------------------
template <bool USE_XBUF>
__global__ __launch_bounds__(128)
void bkt_scan(const int* __restrict__ corr, const int* __restrict__ kcv,
              const int* __restrict__ prob, const float* __restrict__ FM,
              const float* __restrict__ trans, const float* __restrict__ olp,
              const float* __restrict__ olk, const float* __restrict__ initl,
              const float* __restrict__ lrw, const float* __restrict__ lrb,
              const float* __restrict__ xbuf, float* __restrict__ out) {
    __shared__ float alpha[4 * NKC * 2 * NA];   // 41.6 KB
    __shared__ float olk_s[NKC * 4];
    __shared__ float logt_s[NKC * 4];
    __shared__ float la0_s[NKC * 2];

    const int tid = threadIdx.x;
    for (int i = tid; i < NKC * 4; i += 128) olk_s[i] = olk[i];
    for (int c = tid; c < NKC; c += 128) {
        float i0 = initl[c * 2], i1 = initl[c * 2 + 1];
        float l = lse2(i0, i1);
        la0_s[c * 2] = i0 - l;
        la0_s[c * 2 + 1] = i1 - l;
        // trans_logits[c][i_new][j_old]; log_softmax over i_new (per column j)
        float t00 = trans[c * 4 + 0], t01 = trans[c * 4 + 1];
        float t10 = trans[c * 4 + 2], t11 = trans[c * 4 + 3];
        float l0 = lse2(t00, t10);
        float l1 = lse2(t01, t11);
        logt_s[c * 4 + 0] = t00 - l0;
        logt_s[c * 4 + 1] = t01 - l1;
        logt_s[c * 4 + 2] = t10 - l0;
        logt_s[c * 4 + 3] = t11 - l1;
    }
    __syncthreads();
    for (int i = tid; i < 4 * NKC * 2 * NA; i += 128) {
        int r = i % (NKC * 2 * NA);
        int c = r / (2 * NA);
        int s = (r % (2 * NA)) / NA;
        alpha[i] = la0_s[c * 2 + s];
    }
    __syncthreads();

    const int wave = tid >> 5, lane = tid & 31;
    const int b = blockIdx.x * 4 + wave;
    const long bT = (long)b * TT;
    float* myA = alpha + wave * (NKC * 2 * NA);
    const int aa = (lane < NA) ? lane : (NA - 1);
    const float abil = -3.0f + 0.5f * (float)aa;    // linspace(-3,3,13)
    float logab = -2.5649493574615367f;             // -log(13)
    const float NEGI = -1.0e30f;

    for (int t = 0; t < TT; ++t) {
        const long idx = bT + t;
        const int obs = corr[idx];
        const int c = kcv[idx];
        const int p = prob[idx];

        float x0, x1;
        if (USE_XBUF) {
            v2f xv = *(const v2f*)(xbuf + idx * 2);
            x0 = xv.x; x1 = xv.y;
        } else {
            const float* fr = FM + idx * 200;
            float p0 = 0.f, p1 = 0.f;
            for (int f = lane; f < 200; f += 32) {
                float fv = fr[f];
                p0 = fmaf(fv, lrw[f], p0);
                p1 = fmaf(fv, lrw[200 + f], p1);
            }
            for (int off = 16; off; off >>= 1) {
                p0 += __shfl_xor(p0, off, 32);
                p1 += __shfl_xor(p1, off, 32);
            }
            x0 = p0 + lrb[0];
            x1 = p1 + lrb[1];
        }

        const float4 op4 = *(const float4*)(olp + (long)p * 4);  // [s][o]
        // logit[s][o] = olp + olk + abilities[o][a] + olfv[s][o]
        float l00 = op4.x + olk_s[c * 4 + 0] - abil + x0;
        float l01 = op4.y + olk_s[c * 4 + 1] + abil - x0;
        float l10 = op4.z + olk_s[c * 4 + 2] - abil + x1;
        float l11 = op4.w + olk_s[c * 4 + 3] + abil - x1;
        // log_softmax over o
        float z0 = lse2(l00, l01), z1 = lse2(l10, l11);
        float lo00 = l00 - z0, lo01 = l01 - z0;
        float lo10 = l10 - z1, lo11 = l11 - z1;

        float la0v = myA[c * 26 + aa];
        float la1v = myA[c * 26 + 13 + aa];

        float lpgb0 = lse2(lo00 + la0v, lo10 + la1v);
        float lpgb1 = lse2(lo01 + la0v, lo11 + la1v);
        float nrm = lse2(lpgb0, lpgb1);
        lpgb0 -= nrm; lpgb1 -= nrm;

        float lps0 = obs ? lo01 : lo00;
        float lps1 = obs ? lo11 : lo10;
        float nla0 = lse2(lps0 + la0v + logt_s[c * 4 + 0],
                          lps1 + la1v + logt_s[c * 4 + 1]);
        float nla1 = lse2(lps0 + la0v + logt_s[c * 4 + 2],
                          lps1 + la1v + logt_s[c * 4 + 3]);
        if (lane < NA) {
            myA[c * 26 + lane] = nla0;
            myA[c * 26 + 13 + lane] = nla1;
        }

        // out[o] = logsumexp_a(log_ability + lpgb[o]), lanes 0-12 active
        float o0, o1;
        {
            float v = (lane < NA) ? (logab + lpgb0) : NEGI;
            float m2 = v;
            for (int off = 8; off; off >>= 1) m2 = fmaxf(m2, __shfl_xor(m2, off, 16));
            float e = __expf(v - m2), s = e;
            for (int off = 8; off; off >>= 1) s += __shfl_xor(s, off, 16);
            o0 = m2 + __logf(s);
        }
        {
            float v = (lane < NA) ? (logab + lpgb1) : NEGI;
            float m2 = v;
            for (int off = 8; off; off >>= 1) m2 = fmaxf(m2, __shfl_xor(m2, off, 16));
            float e = __expf(v - m2), s = e;
            for (int off = 8; off; off >>= 1) s += __shfl_xor(s, off, 16);
            o1 = m2 + __logf(s);
        }
        float onrm = lse2(o0, o1);
        if (lane == 0) {
            v2f ov;
            ov.x = o0 - onrm;
            ov.y = o1 - onrm;
            *(v2f*)(out + idx * 2) = ov;
        }
        logab += obs ? lpgb1 : lpgb0;
    }
}

// ---------------------------------------------------------------------------
extern "C" void kernel_launch(void* const* d_in, const int* in_sizes, int n_in,
                              void* d_out, int out_size, void* d_ws, size_t ws_size,
                              hipStream_t stream) {
    (void)in_sizes; (void)n_in; (void)out_size;
    const int* corr = (const int*)d_in[0];
    const int* kcv  = (const int*)d_in[1];
    const int* prob = (const int*)d_in[2];
    const float* FM    = (const float*)d_in[3];
    const float* trans = (const float*)d_in[4];
    const float* olp   = (const float*)d_in[5];
    const float* olk   = (const float*)d_in[6];
    const float* initl = (const float*)d_in[7];
    const float* lrw   = (const float*)d_in[8];
    const float* lrb   = (const float*)d_in[9];
    float* out = (float*)d_out;

    const size_t xbytes = (size_t)BB * TT * 2 * sizeof(float);  // 1.024 MB
    if (ws_size >= xbytes) {
        float* xbuf = (float*)d_ws;
        hipLaunchKernelGGL(gemm_x_wmma, dim3(2000), dim3(128), 0, stream,
                           FM, lrw, lrb, xbuf);
        hipLaunchKernelGGL((bkt_scan<true>), dim3(64), dim3(128), 0, stream,
                           corr, kcv, prob, FM, trans, olp, olk, initl,
                           lrw, lrb, xbuf, out);
    } else {
        hipLaunchKernelGGL((bkt_scan<false>), dim3(64), dim3(128), 0, stream,
                           corr, kcv, prob, FM, trans, olp, olk, initl,
                           lrw, lrb, (const float*)nullptr, out);
    }
}